// SAGERecommender_6897717477582
// MI455X (gfx1250) — compile-verified
//
#include <hip/hip_runtime.h>
#include <hip/hip_bf16.h>

typedef float v2f __attribute__((ext_vector_type(2)));
typedef float v8f __attribute__((ext_vector_type(8)));

#define NN 100000
#define NE 1250000
#define DF 64

// ---------------- zero fill ----------------
__global__ void zero_f32(float* __restrict__ p, int n) {
  int i = blockIdx.x * blockDim.x + threadIdx.x;
  int stride = gridDim.x * blockDim.x;
  for (; i < n; i += stride) p[i] = 0.0f;
}

// ---------------- degree count ----------------
__global__ void degree_kernel(const long long* __restrict__ ei,
                              float* __restrict__ cnt) {
  int e = blockIdx.x * blockDim.x + threadIdx.x;
  if (e < NE) {
    long long d = ei[NE + e];
    atomicAdd(&cnt[d], 1.0f);
  }
}

// ---------------- message scatter-add: agg[dst] += xin[src] ----------------
// 256 threads = 4 edges x 64 features; gathers are 256B-contiguous per edge.
__global__ void scatter_kernel(const long long* __restrict__ ei,
                               const float* __restrict__ xin,
                               float* __restrict__ agg) {
  int t = threadIdx.x;
  int e = blockIdx.x * 4 + (t >> 6);
  int f = t & 63;
  if (e < NE) {
    long long s = ei[e];
    long long d = ei[NE + e];
    float v = xin[s * DF + f];
    atomicAdd(&agg[d * DF + f], v);
  }
}

// ---------------- fused mean + dual GEMM + bias (+relu), fp32 WMMA ----------
// One wave32 per 16-node tile: out[16x64] = mean[16x64]*Wl^T + xin[16x64]*Wr^T + b
// A 16x4 f32 frag: lane l holds A[l%16][2*(l>>4)+v]      (ISA 7.12.2)
// B 4x16 f32 frag: lane l holds B[2*(l>>4)+v][l%16] = W[l%16 (+16*nt)][k]
// C 16x16 f32:     VGPR v holds row v+8*(l>>4), col l%16
__global__ void __launch_bounds__(128)
sage_wmma_kernel(const float* __restrict__ xin,
                 const float* __restrict__ agg,
                 const float* __restrict__ cnt,
                 const float* __restrict__ Wl,   // [64][64] row-major (out,in)
                 const float* __restrict__ Wr,   // [64][64]
                 const float* __restrict__ bias, // [64]
                 float* __restrict__ out,
                 int relu) {
  const int lane = threadIdx.x & 31;
  const int wave = threadIdx.x >> 5;
  const int tile = blockIdx.x * 4 + wave;       // wave-uniform
  if (tile >= NN / 16) return;                  // uniform exit: EXEC stays full

  const int base = tile * 16;
  const int m    = lane & 15;
  const int hk   = lane >> 4;                   // K-pair selector
  const int row  = base + m;

  const float inv = 1.0f / fmaxf(cnt[row], 1.0f);

  v8f c[4];
  #pragma unroll
  for (int nt = 0; nt < 4; ++nt) c[nt] = (v8f){};

  #pragma unroll
  for (int kk = 0; kk < 16; ++kk) {             // K = 64 in steps of 4
    const int k0 = kk * 4 + hk * 2;

    v2f a1, a2;                                  // A1 = mean row, A2 = xin row
    a1.x = agg[row * DF + k0]     * inv;
    a1.y = agg[row * DF + k0 + 1] * inv;
    a2.x = xin[row * DF + k0];
    a2.y = xin[row * DF + k0 + 1];

    #pragma unroll
    for (int nt = 0; nt < 4; ++nt) {
      const int col = nt * 16 + m;
      v2f bl, br;                                // B = W^T  ->  B[k][col] = W[col][k]
      bl.x = Wl[col * DF + k0];
      bl.y = Wl[col * DF + k0 + 1];
      br.x = Wr[col * DF + k0];
      br.y = Wr[col * DF + k0 + 1];

      c[nt] = __builtin_amdgcn_wmma_f32_16x16x4_f32(
          false, a1, false, bl, (short)0, c[nt], false, false);
      c[nt] = __builtin_amdgcn_wmma_f32_16x16x4_f32(
          false, a2, false, br, (short)0, c[nt], false, false);
    }
  }

  // epilogue: + bias, optional relu, store
  #pragma unroll
  for (int nt = 0; nt < 4; ++nt) {
    const int col = nt * 16 + m;
    const float b = bias[col];
    #pragma unroll
    for (int v = 0; v < 8; ++v) {
      const int mrow = v + hk * 8;
      float val = c[nt][v] + b;
      if (relu) val = fmaxf(val, 0.0f);
      out[(size_t)(base + mrow) * DF + col] = val;
    }
  }
}

extern "C" void kernel_launch(void* const* d_in, const int* in_sizes, int n_in,
                              void* d_out, int out_size, void* d_ws, size_t ws_size,
                              hipStream_t stream) {
  const float*     x   = (const float*)d_in[0];
  const long long* ei  = (const long long*)d_in[1];  // int64 [2, E]
  const float*     W1l = (const float*)d_in[2];
  const float*     b1  = (const float*)d_in[3];
  const float*     W1r = (const float*)d_in[4];
  const float*     W2l = (const float*)d_in[5];
  const float*     b2  = (const float*)d_in[6];
  const float*     W2r = (const float*)d_in[7];
  float* out = (float*)d_out;

  float* agg = (float*)d_ws;                    // N*64 f32
  float* h   = agg + (size_t)NN * DF;           // N*64 f32
  float* cnt = h + (size_t)NN * DF;             // N f32

  const int ZB = 256;
  // ---- layer 1 ----
  zero_f32<<<2048, ZB, 0, stream>>>(agg, NN * DF);
  zero_f32<<<256,  ZB, 0, stream>>>(cnt, NN);
  degree_kernel<<<(NE + 255) / 256, 256, 0, stream>>>(ei, cnt);
  scatter_kernel<<<(NE + 3) / 4, 256, 0, stream>>>(ei, x, agg);
  sage_wmma_kernel<<<(NN / 16 + 3) / 4, 128, 0, stream>>>(
      x, agg, cnt, W1l, W1r, b1, h, /*relu=*/1);

  // ---- layer 2 ----
  zero_f32<<<2048, ZB, 0, stream>>>(agg, NN * DF);
  scatter_kernel<<<(NE + 3) / 4, 256, 0, stream>>>(ei, h, agg);
  sage_wmma_kernel<<<(NN / 16 + 3) / 4, 128, 0, stream>>>(
      h, agg, cnt, W2l, W2r, b2, out, /*relu=*/0);
}